// SlidingWindowCmn_9139690406348
// MI455X (gfx1250) — compile-verified
//
#include <hip/hip_runtime.h>

// ---------------------------------------------------------------------------
// Sliding-window CMN, (B=64, T=8192, F=128) f32, CMN_WINDOW=600, MIN=100,
// CENTER=False, NORM_VARS=True.
// Memory-bound (~23us floor at 23.3 TB/s). Strategy: per-wave sequential scan
// along T, streaming input through an LDS ring via gfx1250 async-to-LDS DMA
// (ASYNCcnt pipeline, 48-row lookahead), lag-601 re-read from L2, NT stores
// for the output stream to preserve L2 residency of the input.
// ---------------------------------------------------------------------------

typedef float vf4 __attribute__((ext_vector_type(4)));
typedef int   v4i __attribute__((ext_vector_type(4)));

#define T_DIM   8192
#define F_DIM   128
#define B_DIM   64
#define NCHUNK  8
#define TC      (T_DIM / NCHUNK)   // 1024 time steps per chunk
#define LAG     601                // steady window = [t-600, t+1) -> drop x[t-601]
#define MINW    100
#define RING    64                 // LDS ring rows (power of two), >= LOOK+1
#define LOOK    48                 // async lookahead rows (< 63 ASYNCcnt cap)

#if defined(__gfx1250__) && __has_builtin(__builtin_amdgcn_global_load_async_to_lds_b128)
#define HAVE_ASYNC 1
// Pointee-typed, address-space-qualified pointers for the async builtin:
typedef __attribute__((address_space(1))) v4i* gbl_v4i_p;   // global (AS1)
typedef __attribute__((address_space(3))) v4i* lds_v4i_p;   // LDS    (AS3)
#else
#define HAVE_ASYNC 0
#endif

#if defined(__gfx1250__) && __has_builtin(__builtin_amdgcn_s_wait_asynccnt)
#define WAIT_ASYNC(n) __builtin_amdgcn_s_wait_asynccnt(n)
#elif HAVE_ASYNC
#define WAIT_ASYNC(n) asm volatile("s_wait_asynccnt %0" :: "i"(n) : "memory")
#else
#define WAIT_ASYNC(n) do {} while (0)
#endif

__device__ __forceinline__ void issue_row(const float* src, float* ringp, int t) {
#if HAVE_ASYNC
    // DMA one 16B element per lane: wave covers a full 512B feature row.
    __builtin_amdgcn_global_load_async_to_lds_b128(
        (gbl_v4i_p)(unsigned long long)(src + (size_t)t * F_DIM),
        (lds_v4i_p)(unsigned int)(unsigned long long)
            (ringp + (size_t)(t & (RING - 1)) * F_DIM),
        0, 0);
#else
    *(vf4*)(ringp + (size_t)(t & (RING - 1)) * F_DIM) =
        *(const vf4*)(src + (size_t)t * F_DIM);
#endif
}

__device__ __forceinline__ vf4 read_row(const float* ringp, int t) {
    // Oldest in-flight row must be complete: allow LOOK-1 outstanding.
    WAIT_ASYNC(LOOK - 1);
#if HAVE_ASYNC
    asm volatile("" ::: "memory");
#endif
    return *(const vf4*)(ringp + (size_t)(t & (RING - 1)) * F_DIM);
}

__device__ __forceinline__ vf4 v_rsq4(vf4 v) {
    vf4 r;
    r.x = __builtin_amdgcn_rsqf(v.x);
    r.y = __builtin_amdgcn_rsqf(v.y);
    r.z = __builtin_amdgcn_rsqf(v.z);
    r.w = __builtin_amdgcn_rsqf(v.w);
    return r;
}

__device__ __forceinline__ void nt_store4(float* p, vf4 v) {
    __builtin_nontemporal_store(v, (vf4*)p);
}

__global__ __launch_bounds__(32)
void swcmn_kernel(const float* __restrict__ in, float* __restrict__ out) {
    __shared__ float ring[RING * F_DIM];   // 32 KB

    const int chunk = blockIdx.x;
    const int b     = blockIdx.y;
    const int f0    = threadIdx.x * 4;     // 4 consecutive features per lane

    const float* src  = in  + (size_t)b * T_DIM * F_DIM + f0;
    float*       dst  = out + (size_t)b * T_DIM * F_DIM + f0;
    float*       ringp = &ring[f0];

    const int t0    = chunk * TC;
    const int t_beg = (chunk == 0) ? 0 : (t0 - LAG);  // halo warm-up for k>=1
    const int t_end = t0 + TC;

    // ---- prologue: fill the async pipeline ----
    int ti = t_beg;
#pragma unroll 1
    for (int i = 0; i < LOOK; ++i) {
        if (ti < t_end) issue_row(src, ringp, ti);
        ++ti;
    }

    vf4 s = {0.f, 0.f, 0.f, 0.f};
    vf4 q = {0.f, 0.f, 0.f, 0.f};

    if (chunk == 0) {
        // Phase A: accumulate stats over [0, 100)
#pragma unroll 1
        for (int t = 0; t < MINW; ++t) {
            vf4 x = read_row(ringp, t);
            s += x; q += x * x;
            if (ti < t_end) issue_row(src, ringp, ti);
            ++ti;
        }
        // Phase B: first 100 frames share fixed stats; re-read x (L1/L2 hot)
        {
            const float invn = 1.0f / (float)MINW;
            vf4 mean = s * invn;
            vf4 rs   = v_rsq4(q * invn - mean * mean);
#pragma unroll 4
            for (int t = 0; t < MINW; ++t) {
                vf4 x = *(const vf4*)(src + (size_t)t * F_DIM);
                nt_store4(dst + (size_t)t * F_DIM, (x - mean) * rs);
            }
        }
        // Phase C: growing window [0, t+1), t = 100..600
#pragma unroll 1
        for (int t = MINW; t < LAG; ++t) {
            vf4 x = read_row(ringp, t);
            s += x; q += x * x;
            const float invn = __builtin_amdgcn_rcpf((float)(t + 1));
            vf4 mean = s * invn;
            vf4 var  = q * invn - mean * mean;
            nt_store4(dst + (size_t)t * F_DIM, (x - mean) * v_rsq4(var));
            if (ti < t_end) issue_row(src, ringp, ti);
            ++ti;
        }
    } else {
        // Halo warm-up: accumulate [t0-601, t0) through the same pipeline
#pragma unroll 1
        for (int t = t_beg; t < t0; ++t) {
            vf4 x = read_row(ringp, t);
            s += x; q += x * x;
            if (ti < t_end) issue_row(src, ringp, ti);
            ++ti;
        }
    }

    // ---- steady state: fixed 601-frame window ----
    const float inv601 = 1.0f / 601.0f;
    const int t_st = (chunk == 0) ? LAG : t0;
#pragma unroll 2
    for (int t = t_st; t < t_end; ++t) {
        vf4 x  = read_row(ringp, t);
        vf4 xo = *(const vf4*)(src + (size_t)(t - LAG) * F_DIM);  // L2-resident lag read
        s += x - xo;
        q += x * x - xo * xo;
        vf4 mean = s * inv601;
        vf4 var  = q * inv601 - mean * mean;
        nt_store4(dst + (size_t)t * F_DIM, (x - mean) * v_rsq4(var));
        if (ti < t_end) issue_row(src, ringp, ti);
        ++ti;
    }
}

extern "C" void kernel_launch(void* const* d_in, const int* in_sizes, int n_in,
                              void* d_out, int out_size, void* d_ws, size_t ws_size,
                              hipStream_t stream) {
    (void)in_sizes; (void)n_in; (void)d_ws; (void)ws_size; (void)out_size;
    const float* in  = (const float*)d_in[0];
    float*       out = (float*)d_out;
    dim3 grid(NCHUNK, B_DIM);   // 8 T-chunks x 64 batches, 1 wave each
    swcmn_kernel<<<grid, 32, 0, stream>>>(in, out);
}